// SSDBoxHead_51161650430689
// MI455X (gfx1250) — compile-verified
//
#include <hip/hip_runtime.h>
#include <hip/hip_bf16.h>
#include <stdint.h>

// Problem constants (from reference setup_inputs)
#define BATCH   32
#define NPRIORS 24564
#define NCLS    81
#define ROWS    (BATCH * NPRIORS)

#define TILE      256   // priors per decode block
#define B_PER_BLK 8     // batches handled per decode block (gridDim.y = BATCH/B_PER_BLK)

typedef float        f32x4 __attribute__((ext_vector_type(4)));
typedef unsigned int u32x4 __attribute__((ext_vector_type(4)));
typedef int          i32x4 __attribute__((ext_vector_type(4)));
typedef int          i32x8 __attribute__((ext_vector_type(8)));

#if __has_builtin(__builtin_amdgcn_tensor_load_to_lds) && __has_builtin(__builtin_amdgcn_s_wait_tensorcnt)
#define HAVE_TDM 1
#endif

// ---------------------------------------------------------------------------
// Kernel 1: row softmax over C=81. One wave32 per row; lane i owns elements
// i, i+32, i+64 (third predicated to lanes 0..16). NT loads/stores: this
// stream (2 x 255 MB) exceeds the 192 MB L2, so bypassing retention is right.
// ---------------------------------------------------------------------------
__global__ __launch_bounds__(256) void softmax_rows_kernel(
    const float* __restrict__ logits, float* __restrict__ scores, int nrows) {
  const int lane = threadIdx.x & 31;
  const int wave = threadIdx.x >> 5;
  const int row  = blockIdx.x * 8 + wave;   // 8 waves per block
  if (row >= nrows) return;

  const long long base = (long long)row * NCLS;
  const float* p = logits + base;

  const bool has3 = lane < (NCLS - 64);     // lanes 0..16 own a third element
  float x0 = __builtin_nontemporal_load(p + lane);
  float x1 = __builtin_nontemporal_load(p + 32 + lane);
  float x2 = has3 ? __builtin_nontemporal_load(p + 64 + lane) : -__builtin_inff();

  // wave32 max-reduce
  float m = fmaxf(fmaxf(x0, x1), x2);
#pragma unroll
  for (int off = 16; off > 0; off >>= 1)
    m = fmaxf(m, __shfl_xor(m, off, 32));

  float e0 = __expf(x0 - m);
  float e1 = __expf(x1 - m);
  float e2 = has3 ? __expf(x2 - m) : 0.0f;

  // wave32 sum-reduce
  float s = e0 + e1 + e2;
#pragma unroll
  for (int off = 16; off > 0; off >>= 1)
    s += __shfl_xor(s, off, 32);

  const float inv = 1.0f / s;
  float* q = scores + base;
  __builtin_nontemporal_store(e0 * inv, q + lane);
  __builtin_nontemporal_store(e1 * inv, q + 32 + lane);
  if (has3) __builtin_nontemporal_store(e2 * inv, q + 64 + lane);
}

// ---------------------------------------------------------------------------
// Kernel 2: bbox decode. Stage a TILE x float4 prior tile into LDS via the
// Tensor Data Mover (async DMA, TENSORcnt), then each thread decodes its
// prior for B_PER_BLK batches (LDS reuse) with 16B vector NT traffic.
// ---------------------------------------------------------------------------
__global__ __launch_bounds__(256) void decode_boxes_kernel(
    const float* __restrict__ bbox, const float* __restrict__ priors,
    float* __restrict__ boxes) {
  __shared__ f32x4 tile[TILE];
  const int n0    = blockIdx.x * TILE;
  const int count = min(TILE, NPRIORS - n0);
  const int tid   = threadIdx.x;

#ifdef HAVE_TDM
  if (tid < 32) {  // wave 0 issues the TDM descriptor (EXEC ignored by tensor op)
    const uint64_t ga    = (uint64_t)(uintptr_t)(priors + (size_t)n0 * 4);
    // Generic LDS pointer: low 32 bits are the workgroup-relative LDS offset.
    const unsigned ldsa  = (unsigned)(uintptr_t)&tile[0];
    const unsigned nelem = (unsigned)count * 4u;  // 4-byte elements in the tile

    // D# group 0: count=1 | lds_addr | global_addr[56:0] | type=2
    u32x4 g0 = {1u, ldsa, (unsigned)ga,
                (unsigned)((ga >> 32) & 0x1FFFFFFu) | 0x80000000u};
    // D# group 1: wg_mask=0, data_size=2 (4B); 1-D tile of `nelem` elements.
    //   tensor_dim0 = nelem  (bits 79:48)
    //   tensor_dim1 = 1      (bits 111:80)
    //   tile_dim0   = nelem  (bits 127:112)
    //   tile_dim1   = 1      (bits 143:128), tile_dim2 = 0
    //   tensor_dim0_stride = nelem (bits 207:160), dim1_stride = 0
    i32x8 g1 = {(int)(2u << 16),
                (int)((nelem & 0xFFFFu) << 16),
                (int)(((nelem >> 16) & 0xFFFFu) | (1u << 16)),
                (int)((nelem & 0xFFFFu) << 16),
                1,
                (int)nelem,
                0, 0};
    i32x4 gz  = {0, 0, 0, 0};
#if __clang_major__ >= 23
    i32x8 gz8 = {0, 0, 0, 0, 0, 0, 0, 0};
    __builtin_amdgcn_tensor_load_to_lds(g0, g1, gz, gz, gz8, 0);
#else
    __builtin_amdgcn_tensor_load_to_lds(g0, g1, gz, gz, 0);
#endif
    __builtin_amdgcn_s_wait_tensorcnt(0);
  }
  __syncthreads();
#else
  if (tid < count) tile[tid] = ((const f32x4*)priors)[n0 + tid];
  __syncthreads();
#endif

  if (tid < count) {
    const int n = n0 + tid;
    const f32x4 pr = tile[tid];              // (px, py, pw, ph)
    const f32x4* bp = (const f32x4*)bbox;
    f32x4* out = (f32x4*)boxes;
    const int b0 = blockIdx.y * B_PER_BLK;
#pragma unroll 4
    for (int b = b0; b < b0 + B_PER_BLK; ++b) {
      const size_t idx = (size_t)b * NPRIORS + n;
      f32x4 t = __builtin_nontemporal_load(bp + idx);
      const float cx = t.x * 0.1f * pr.z + pr.x;
      const float cy = t.y * 0.1f * pr.w + pr.y;
      const float hw = __expf(t.z * 0.2f) * pr.z * 0.5f;
      const float hh = __expf(t.w * 0.2f) * pr.w * 0.5f;
      f32x4 o = {cx - hw, cy - hh, cx + hw, cy + hh};
      __builtin_nontemporal_store(o, out + idx);
    }
  }
}

// ---------------------------------------------------------------------------
extern "C" void kernel_launch(void* const* d_in, const int* in_sizes, int n_in,
                              void* d_out, int out_size, void* d_ws, size_t ws_size,
                              hipStream_t stream) {
  (void)in_sizes; (void)n_in; (void)d_ws; (void)ws_size; (void)out_size;
  const float* cls    = (const float*)d_in[0];  // (32, 24564, 81) f32
  const float* bbox   = (const float*)d_in[1];  // (32, 24564, 4)  f32
  const float* priors = (const float*)d_in[2];  // (24564, 4)      f32

  float* scores = (float*)d_out;                               // first output
  float* boxes  = (float*)d_out + (size_t)ROWS * NCLS;         // second output

  // Softmax: 786048 rows, 8 rows (waves) per 256-thread block.
  softmax_rows_kernel<<<(ROWS + 7) / 8, 256, 0, stream>>>(cls, scores, ROWS);

  // Decode: (ceil(N/TILE), BATCH/B_PER_BLK) grid.
  dim3 grid((NPRIORS + TILE - 1) / TILE, BATCH / B_PER_BLK);
  decode_boxes_kernel<<<grid, 256, 0, stream>>>(bbox, priors, boxes);
}